// GravityDecoder_86328842650110
// MI455X (gfx1250) — compile-verified
//
#include <hip/hip_runtime.h>
#include <hip/hip_bf16.h>

typedef __attribute__((ext_vector_type(2))) float v2f;
typedef __attribute__((ext_vector_type(8))) float v8f;

#define GD_EPS 1e-7f
#define GD_D   128

// One wave32 per 16 edges.
//  accD = diff * diff^T  -> diagonal = ||z_i - z_j||^2
//  accM = z_j  * Wbcast  -> every column = z_j . W
__global__ __launch_bounds__(256) void GravityDecoder_edge_kernel(
    const float* __restrict__ z,          // [N,128]
    const int*   __restrict__ edge_index, // [2,E]
    const float* __restrict__ W,          // [128]
    const float* __restrict__ bptr,       // [1]
    float*       __restrict__ out,        // [4*E] logits|prob|m_j|dist2
    int E)
{
    const int lane = threadIdx.x & 31;
    const int col  = lane & 15;   // edge within tile / matrix column
    const int half = lane >> 4;   // K sub-slot (0 -> k+0..1, 1 -> k+2..3)

    const long long tile = (long long)blockIdx.x * (blockDim.x >> 5) + (threadIdx.x >> 5);
    const long long e0 = tile * 16;
    if (e0 >= (long long)E) return;   // wave-uniform: tile id is per-wave

    long long e = e0 + col;
    if (e >= (long long)E) e = (long long)E - 1;   // clamp tail, keep EXEC all-1s

    const long long srcRow = (long long)edge_index[e];              // edge_index[0][e]
    const long long dstRow = (long long)edge_index[(long long)E + e]; // edge_index[1][e]

    const float* pi = z + srcRow * GD_D + half * 2;
    const float* pj = z + dstRow * GD_D + half * 2;
    const float* pw = W + half * 2;

    v8f accD = {0.f, 0.f, 0.f, 0.f, 0.f, 0.f, 0.f, 0.f};
    v8f accM = {0.f, 0.f, 0.f, 0.f, 0.f, 0.f, 0.f, 0.f};

#pragma unroll
    for (int k0 = 0; k0 < GD_D; k0 += 4) {
        v2f ai = *(const v2f*)(pi + k0);   // z_i fragment (A layout)
        v2f bj = *(const v2f*)(pj + k0);   // z_j fragment (A and B^T layouts coincide)
        v2f wv = *(const v2f*)(pw + k0);   // W broadcast fragment (B layout, col-invariant)
        v2f d  = ai - bj;
        // D = A*B + C, fp32 16x16x4
        accD = __builtin_amdgcn_wmma_f32_16x16x4_f32(
                   false, d,  false, d,  (short)0, accD, false, false);
        accM = __builtin_amdgcn_wmma_f32_16x16x4_f32(
                   false, bj, false, wv, (short)0, accM, false, false);
    }

    // Diagonal element for edge idx lives at:
    //   e_loc 0..7  -> lane e_loc,      vgpr r = e_loc
    //   e_loc 8..15 -> lane e_loc + 16, vgpr r = e_loc - 8
    const int idx = col - half * 8;       // valid in [0,8) only on diagonal holders
    float dD = accD[0];
    float dM = accM[0];
#pragma unroll
    for (int r = 1; r < 8; ++r) {
        if (idx == r) { dD = accD[r]; dM = accM[r]; }
    }

    // Gather edge e_loc's diagonal value into lane e_loc (backward permute, byte index).
    const int src_lane = (col < 8) ? col : (col + 16);
    const float dist2 = __int_as_float(
        __builtin_amdgcn_ds_bpermute(src_lane << 2, __float_as_int(dD))) + GD_EPS;
    const float mj = __int_as_float(
        __builtin_amdgcn_ds_bpermute(src_lane << 2, __float_as_int(dM))) + bptr[0];

    if (lane < 16) {
        const long long eo = e0 + lane;
        if (eo < (long long)E) {
            const float logits = mj - __logf(dist2);
            const float prob   = 1.0f / (1.0f + __expf(-logits));
            out[eo]                      = logits;
            out[(long long)E       + eo] = prob;
            out[(long long)E * 2ll + eo] = mj;
            out[(long long)E * 3ll + eo] = dist2;
        }
    }
}

extern "C" void kernel_launch(void* const* d_in, const int* in_sizes, int n_in,
                              void* d_out, int out_size, void* d_ws, size_t ws_size,
                              hipStream_t stream) {
    const float* z  = (const float*)d_in[0];
    const int*   ei = (const int*)  d_in[1];
    const float* W  = (const float*)d_in[2];
    const float* b  = (const float*)d_in[3];
    float* out = (float*)d_out;

    const int E = in_sizes[1] / 2;                 // edge_index is [2,E]
    const int tiles = (E + 15) / 16;               // one wave32 per tile
    const int wavesPerBlock = 256 / 32;
    const int blocks = (tiles + wavesPerBlock - 1) / wavesPerBlock;

    hipLaunchKernelGGL(GravityDecoder_edge_kernel,
                       dim3(blocks), dim3(256), 0, stream,
                       z, ei, W, b, out, E);
}